// IMUPreintegrator_44246753083448
// MI455X (gfx1250) — compile-verified
//
#include <hip/hip_runtime.h>
#include <hip/hip_bf16.h>
#include <math.h>

// ---------------------------------------------------------------------------
// IMU preintegration for MI455X (gfx1250, wave32).
// out = [incre_q (32,8192,4) | vel (32,8192,3) | pos (32,8192,3) | cov (32,9,9)]
//
// Covariance recurrence C' = A C A^T + N is composed associatively:
//   (A2,N2) o (A1,N1) = (A2*A1, A2*N1*A2^T + N2),  C0 = 0  =>  cov = N_total.
// 9x9 matrices are zero-padded to 16x16 and multiplied with chained
// V_WMMA_F32_16X16X4_F32 (FP32 needed: cov entries ~1e-10 underflow FP16).
// ---------------------------------------------------------------------------

#define B_DIM 32
#define F_DIM 8192
#define GRAV 9.81007f
#define GYRO_COV_C (0.0032f * 0.0032f)
#define ACC_COV_C  (0.08f * 0.08f)

typedef float v2f __attribute__((ext_vector_type(2)));
typedef float v8f __attribute__((ext_vector_type(8)));

// Compiler-only reordering fence; per-wave LDS ops are in-order in HW (DScnt).
#define LDS_FENCE() asm volatile("" ::: "memory")

struct Q4 { float x, y, z, w; };

__device__ __forceinline__ Q4 qmul(const Q4 a, const Q4 b) {
  Q4 r;
  r.x = a.w * b.x + a.x * b.w + a.y * b.z - a.z * b.y;
  r.y = a.w * b.y - a.x * b.z + a.y * b.w + a.z * b.x;
  r.z = a.w * b.z + a.x * b.y - a.y * b.x + a.z * b.w;
  r.w = a.w * b.w - a.x * b.x - a.y * b.y - a.z * b.z;
  return r;
}

__device__ __forceinline__ Q4 qexp3(float wx, float wy, float wz) {
  float th2 = wx * wx + wy * wy + wz * wz;
  float s, c;
  if (th2 < 1e-8f) {
    s = 0.5f - th2 * (1.0f / 48.0f);
    c = 1.0f - th2 * 0.125f;
  } else {
    float th = sqrtf(th2);
    s = sinf(0.5f * th) / th;
    c = cosf(0.5f * th);
  }
  Q4 q; q.x = wx * s; q.y = wy * s; q.z = wz * s; q.w = c;
  return q;
}

// Row r of the rotation matrix of unit quaternion q (xyzw).
__device__ __forceinline__ void rotrow(const Q4 q, int r, float* m) {
  float x = q.x, y = q.y, z = q.z, w = q.w;
  if (r == 0) {
    m[0] = 1.0f - 2.0f * (y * y + z * z);
    m[1] = 2.0f * (x * y - z * w);
    m[2] = 2.0f * (x * z + y * w);
  } else if (r == 1) {
    m[0] = 2.0f * (x * y + z * w);
    m[1] = 1.0f - 2.0f * (x * x + z * z);
    m[2] = 2.0f * (y * z - x * w);
  } else {
    m[0] = 2.0f * (x * z - y * w);
    m[1] = 2.0f * (y * z + x * w);
    m[2] = 1.0f - 2.0f * (x * x + y * y);
  }
}

// K = skew(w) entry (i,j)
__device__ __forceinline__ float skew_ent(int i, int j, float wx, float wy, float wz) {
  if (i == 0) return (j == 1) ? -wz : ((j == 2) ? wy : 0.0f);
  if (i == 1) return (j == 0) ? wz : ((j == 2) ? -wx : 0.0f);
  return (j == 0) ? -wy : ((j == 1) ? wx : 0.0f);
}

// ---------------------------------------------------------------------------
// Kernel 1: quaternion scan + velocity/position cumsums. One block per batch.
// ---------------------------------------------------------------------------
__global__ __launch_bounds__(256) void k_integrate(
    const float* __restrict__ dt, const float* __restrict__ gyro,
    const float* __restrict__ acc, const float* __restrict__ pos0,
    const float* __restrict__ vel0,
    float* __restrict__ quat_out, float* __restrict__ vel_out,
    float* __restrict__ pos_out) {
  __shared__ float sq[256][4];
  __shared__ float sv[256][3];
  const int b = blockIdx.x, t = threadIdx.x;
  const int CH = F_DIM / 256;  // 32 steps per thread
  const float* dtb = dt + (size_t)b * F_DIM;
  const float* gb  = gyro + (size_t)b * F_DIM * 3;
  const float* ab  = acc + (size_t)b * F_DIM * 3;
  float* qo = quat_out + (size_t)b * F_DIM * 4;
  float* vo = vel_out + (size_t)b * F_DIM * 3;
  float* po = pos_out + (size_t)b * F_DIM * 3;
  const int k0 = t * CH;

  // --- phase 1: thread-local inclusive quaternion products ---
  Q4 ql = {0.f, 0.f, 0.f, 1.f};
  for (int s = 0; s < CH; ++s) {
    int k = k0 + s;
    float d = dtb[k];
    Q4 dq = qexp3(gb[3 * k] * d, gb[3 * k + 1] * d, gb[3 * k + 2] * d);
    ql = qmul(ql, dq);
    qo[4 * k] = ql.x; qo[4 * k + 1] = ql.y; qo[4 * k + 2] = ql.z; qo[4 * k + 3] = ql.w;
  }
  sq[t][0] = ql.x; sq[t][1] = ql.y; sq[t][2] = ql.z; sq[t][3] = ql.w;
  __syncthreads();
  // --- block-wide inclusive scan (noncommutative, earlier operand on left) ---
  for (int dd = 1; dd < 256; dd <<= 1) {
    Q4 v = {sq[t][0], sq[t][1], sq[t][2], sq[t][3]};
    Q4 res = v;
    if (t >= dd) {
      Q4 u = {sq[t - dd][0], sq[t - dd][1], sq[t - dd][2], sq[t - dd][3]};
      res = qmul(u, v);
    }
    __syncthreads();
    sq[t][0] = res.x; sq[t][1] = res.y; sq[t][2] = res.z; sq[t][3] = res.w;
    __syncthreads();
  }
  Q4 P = {0.f, 0.f, 0.f, 1.f};
  if (t > 0) { P.x = sq[t - 1][0]; P.y = sq[t - 1][1]; P.z = sq[t - 1][2]; P.w = sq[t - 1][3]; }

  // --- phase 2: fix up quats; compute dv_step into vel_out (staging) ---
  float vs0 = 0.f, vs1 = 0.f, vs2 = 0.f;
  Q4 qprev = P;  // DeltaR_{ik} rotation before first local step
  for (int s = 0; s < CH; ++s) {
    int k = k0 + s;
    Q4 qlk = {qo[4 * k], qo[4 * k + 1], qo[4 * k + 2], qo[4 * k + 3]};
    Q4 qk = qmul(P, qlk);
    qo[4 * k] = qk.x; qo[4 * k + 1] = qk.y; qo[4 * k + 2] = qk.z; qo[4 * k + 3] = qk.w;
    float d = dtb[k];
    float r2[3]; rotrow(qk, 2, r2);  // row 2 of R_incl; a = acc - R^T g
    float a0 = ab[3 * k]     - GRAV * r2[0];
    float a1 = ab[3 * k + 1] - GRAV * r2[1];
    float a2 = ab[3 * k + 2] - GRAV * r2[2];
    float p0[3], p1[3], p2[3];
    rotrow(qprev, 0, p0); rotrow(qprev, 1, p1); rotrow(qprev, 2, p2);
    float dv0 = (p0[0] * a0 + p0[1] * a1 + p0[2] * a2) * d;
    float dv1 = (p1[0] * a0 + p1[1] * a1 + p1[2] * a2) * d;
    float dv2 = (p2[0] * a0 + p2[1] * a1 + p2[2] * a2) * d;
    vo[3 * k] = dv0; vo[3 * k + 1] = dv1; vo[3 * k + 2] = dv2;
    vs0 += dv0; vs1 += dv1; vs2 += dv2;
    qprev = qk;
  }
  sv[t][0] = vs0; sv[t][1] = vs1; sv[t][2] = vs2;
  __syncthreads();
  for (int dd = 1; dd < 256; dd <<= 1) {
    float x0 = sv[t][0], x1 = sv[t][1], x2 = sv[t][2];
    if (t >= dd) { x0 += sv[t - dd][0]; x1 += sv[t - dd][1]; x2 += sv[t - dd][2]; }
    __syncthreads();
    sv[t][0] = x0; sv[t][1] = x1; sv[t][2] = x2;
    __syncthreads();
  }
  float Vp0 = 0.f, Vp1 = 0.f, Vp2 = 0.f;
  if (t > 0) { Vp0 = sv[t - 1][0]; Vp1 = sv[t - 1][1]; Vp2 = sv[t - 1][2]; }
  __syncthreads();

  // --- phase 3: finalize vel; stage pos integrand into pos_out ---
  const float v00 = vel0[b * 3], v01 = vel0[b * 3 + 1], v02 = vel0[b * 3 + 2];
  float run0 = Vp0, run1 = Vp1, run2 = Vp2;
  float ps0 = 0.f, ps1 = 0.f, ps2 = 0.f;
  for (int s = 0; s < CH; ++s) {
    int k = k0 + s;
    float dv0 = vo[3 * k], dv1 = vo[3 * k + 1], dv2 = vo[3 * k + 2];
    run0 += dv0; run1 += dv1; run2 += dv2;
    float d = dtb[k];
    float q0 = (run0 - dv0) * d + 0.5f * dv0 * d;  // dvel_prev*dt + 0.5*dv*dt
    float q1 = (run1 - dv1) * d + 0.5f * dv1 * d;
    float q2 = (run2 - dv2) * d + 0.5f * dv2 * d;
    po[3 * k] = q0; po[3 * k + 1] = q1; po[3 * k + 2] = q2;
    ps0 += q0; ps1 += q1; ps2 += q2;
    vo[3 * k] = v00 + run0; vo[3 * k + 1] = v01 + run1; vo[3 * k + 2] = v02 + run2;
  }
  sv[t][0] = ps0; sv[t][1] = ps1; sv[t][2] = ps2;
  __syncthreads();
  for (int dd = 1; dd < 256; dd <<= 1) {
    float x0 = sv[t][0], x1 = sv[t][1], x2 = sv[t][2];
    if (t >= dd) { x0 += sv[t - dd][0]; x1 += sv[t - dd][1]; x2 += sv[t - dd][2]; }
    __syncthreads();
    sv[t][0] = x0; sv[t][1] = x1; sv[t][2] = x2;
    __syncthreads();
  }
  float Pp0 = 0.f, Pp1 = 0.f, Pp2 = 0.f;
  if (t > 0) { Pp0 = sv[t - 1][0]; Pp1 = sv[t - 1][1]; Pp2 = sv[t - 1][2]; }
  const float p00 = pos0[b * 3], p01 = pos0[b * 3 + 1], p02 = pos0[b * 3 + 2];
  run0 = Pp0; run1 = Pp1; run2 = Pp2;
  for (int s = 0; s < CH; ++s) {
    int k = k0 + s;
    run0 += po[3 * k]; run1 += po[3 * k + 1]; run2 += po[3 * k + 2];
    po[3 * k] = p00 + run0; po[3 * k + 1] = p01 + run1; po[3 * k + 2] = p02 + run2;
  }
}

// ---------------------------------------------------------------------------
// Kernel 2: covariance composition with V_WMMA_F32_16X16X4_F32.
// One block per batch, 8 waves; each wave composes 1024 steps, then wave 0
// sequentially folds the 8 wave results and writes the 9x9 covariance.
// Tiles are row-major 16x16 in LDS; operand layouts (per ISA 7.12.2):
//   A-op (16x4 chunk c): lane L holds A[L%16][4c + 2*(L/16) + v], v=0,1
//   B-op (4x16 chunk c): lane L holds B[4c + 2*(L/16) + v][L%16]
//   C/D : lane L, vgpr r holds D[8*(L/16) + r][L%16]
// Note B-op of M^T reads the same slots as A-op of M -> register reuse.
// ---------------------------------------------------------------------------
#define WMMA4(a, b, c) \
  __builtin_amdgcn_wmma_f32_16x16x4_f32(false, (a), false, (b), (short)0, (c), false, false)

__global__ __launch_bounds__(256) void k_cov(
    const float* __restrict__ dt, const float* __restrict__ gyro,
    const float* __restrict__ acc, const float* __restrict__ quat,
    float* __restrict__ cov_out) {
  __shared__ float tiles[8][5][256];  // 40 KB: A, N, Y, A2, N2 per wave
  const int b = blockIdx.x;
  const int wave = threadIdx.x >> 5;
  const int lane = threadIdx.x & 31;
  float* At  = tiles[wave][0];
  float* Nt  = tiles[wave][1];
  float* Yt  = tiles[wave][2];
  float* A2t = tiles[wave][3];
  float* N2t = tiles[wave][4];

  // init: zero everything, identity accumulator A, static identity blocks of A2
  for (int i = lane; i < 256; i += 32) {
    At[i] = 0.f; Nt[i] = 0.f; Yt[i] = 0.f; A2t[i] = 0.f; N2t[i] = 0.f;
  }
  if (lane < 9) At[lane * 16 + lane] = 1.0f;
  if (lane < 6) { int p = 3 + lane; A2t[p * 16 + p] = 1.0f; }  // rows/cols 3..8 diag
  LDS_FENCE();

  const float* dtb = dt + (size_t)b * F_DIM;
  const float* gb  = gyro + (size_t)b * F_DIM * 3;
  const float* ab  = acc + (size_t)b * F_DIM * 3;
  const float* qb  = quat + (size_t)b * F_DIM * 4;

  const int CH = F_DIM / 8;  // 1024 steps per wave
  const int kbeg = wave * CH;
  const int ln = lane & 15;
  const int hg = lane >> 4;
  const int mb = 8 * hg;

  for (int s = 0; s < CH; ++s) {
    const int k = kbeg + s;
    if ((s & 63) == 0) {  // stage upcoming input cachelines
      int kp = (k + 64 < F_DIM) ? k + 64 : k;
      __builtin_prefetch(dtb + kp, 0, 0);
      __builtin_prefetch(gb + 3 * kp, 0, 0);
      __builtin_prefetch(ab + 3 * kp, 0, 0);
      __builtin_prefetch(qb + 4 * kp, 0, 0);
    }
    const float d = dtb[k];
    const float wx = gb[3 * k] * d, wy = gb[3 * k + 1] * d, wz = gb[3 * k + 2] * d;
    Q4 qc = {qb[4 * k], qb[4 * k + 1], qb[4 * k + 2], qb[4 * k + 3]};
    Q4 qp;
    if (k == 0) { qp.x = 0.f; qp.y = 0.f; qp.z = 0.f; qp.w = 1.f; }
    else { qp.x = qb[4 * k - 4]; qp.y = qb[4 * k - 3]; qp.z = qb[4 * k - 2]; qp.w = qb[4 * k - 1]; }
    float r2[3]; rotrow(qc, 2, r2);
    const float a0 = ab[3 * k]     - GRAV * r2[0];
    const float a1 = ab[3 * k + 1] - GRAV * r2[1];
    const float a2 = ab[3 * k + 2] - GRAV * r2[2];

    // --- build per-step A2 (30 dynamic entries) and N2 (21 entries) ---
    if (lane < 9) {
      int r = lane / 3, c = lane % 3;
      Q4 dq = qexp3(wx, wy, wz);
      float rc[3]; rotrow(dq, c, rc);       // dRT[r][c] = R(dq)[c][r]
      float pr[3]; rotrow(qp, r, pr);       // row r of R_pre
      float Ra;                             // (R_pre * skew(a))[r][c]
      if (c == 0)      Ra =  pr[1] * a2 - pr[2] * a1;
      else if (c == 1) Ra = -pr[0] * a2 + pr[2] * a0;
      else             Ra =  pr[0] * a1 - pr[1] * a0;
      A2t[r * 16 + c] = rc[r];
      A2t[(3 + r) * 16 + c] = -Ra * d;
      A2t[(6 + r) * 16 + c] = -0.5f * Ra * d * d;
    } else if (lane < 18) {
      int j = lane - 9, r = j / 3, c = j % 3;
      float th2 = wx * wx + wy * wy + wz * wz;
      float c1, c2;
      if (th2 < 1e-8f) {
        c1 = 0.5f - th2 * (1.0f / 24.0f);
        c2 = (1.0f / 6.0f) - th2 * (1.0f / 120.0f);
      } else {
        float th = sqrtf(th2);
        c1 = (1.0f - cosf(th)) / th2;
        c2 = (th - sinf(th)) / (th2 * th);
      }
      float wv[3] = {wx, wy, wz};
      float dg = 1.0f - c2 * th2;  // Jr = dg*I - c1*K + c2*w*w^T
      float jr_r[3], jr_c[3];
#pragma unroll
      for (int tt = 0; tt < 3; ++tt) {
        jr_r[tt] = ((r == tt) ? dg : 0.f) - c1 * skew_ent(r, tt, wx, wy, wz) + c2 * wv[r] * wv[tt];
        jr_c[tt] = ((c == tt) ? dg : 0.f) - c1 * skew_ent(c, tt, wx, wy, wz) + c2 * wv[c] * wv[tt];
      }
      float ngg = jr_r[0] * jr_c[0] + jr_r[1] * jr_c[1] + jr_r[2] * jr_c[2];
      N2t[r * 16 + c] = GYRO_COV_C * d * d * ngg;
    } else if (lane < 21) {
      int i = lane - 18;
      A2t[(6 + i) * 16 + (3 + i)] = d;     // I*dt block
      float d2 = d * d;
      N2t[(3 + i) * 16 + (3 + i)] = ACC_COV_C * d2;          // Bv qa Bv^T
      float nvp = 0.5f * ACC_COV_C * d2 * d;                 // Bv qa Bp^T
      N2t[(3 + i) * 16 + (6 + i)] = nvp;
      N2t[(6 + i) * 16 + (3 + i)] = nvp;
      N2t[(6 + i) * 16 + (6 + i)] = 0.25f * ACC_COV_C * d2 * d2;  // Bp qa Bp^T
    }
    LDS_FENCE();

    // --- load operands (a2A also serves as B-operand of A2^T) ---
    v2f a2A[4], accB[4], nccB[4];
#pragma unroll
    for (int c = 0; c < 4; ++c) {
      int kk = 4 * c + 2 * hg;
      a2A[c].x  = A2t[ln * 16 + kk]; a2A[c].y  = A2t[ln * 16 + kk + 1];
      accB[c].x = At[kk * 16 + ln];  accB[c].y = At[(kk + 1) * 16 + ln];
      nccB[c].x = Nt[kk * 16 + ln];  nccB[c].y = Nt[(kk + 1) * 16 + ln];
    }
    v8f n2c;
#pragma unroll
    for (int r = 0; r < 8; ++r) n2c[r] = N2t[(mb + r) * 16 + ln];

    // --- 12 WMMAs: Anew = A2*Aacc ; Y = A2*Nacc ; Nnew = Y*A2^T + N2 ---
    v8f accA = {0.f, 0.f, 0.f, 0.f, 0.f, 0.f, 0.f, 0.f};
    v8f accY = {0.f, 0.f, 0.f, 0.f, 0.f, 0.f, 0.f, 0.f};
#pragma unroll
    for (int c = 0; c < 4; ++c) {
      accA = WMMA4(a2A[c], accB[c], accA);
      accY = WMMA4(a2A[c], nccB[c], accY);
    }
#pragma unroll
    for (int r = 0; r < 8; ++r) {
      At[(mb + r) * 16 + ln] = accA[r];
      Yt[(mb + r) * 16 + ln] = accY[r];
    }
    LDS_FENCE();
    v2f yA[4];
#pragma unroll
    for (int c = 0; c < 4; ++c) {
      int kk = 4 * c + 2 * hg;
      yA[c].x = Yt[ln * 16 + kk]; yA[c].y = Yt[ln * 16 + kk + 1];
    }
    v8f accN = n2c;
#pragma unroll
    for (int c = 0; c < 4; ++c) accN = WMMA4(yA[c], a2A[c], accN);
#pragma unroll
    for (int r = 0; r < 8; ++r) Nt[(mb + r) * 16 + ln] = accN[r];
    LDS_FENCE();
  }

  __syncthreads();

  // --- wave 0 folds the 8 per-wave results (later wave on the left) ---
  if (wave == 0) {
    float* A1 = tiles[0][0];
    float* N1 = tiles[0][1];
    float* Ys = tiles[0][2];
    for (int w2 = 1; w2 < 8; ++w2) {
      const float* A2p = tiles[w2][0];
      const float* N2p = tiles[w2][1];
      v2f a2A[4], a1B[4], n1B[4];
#pragma unroll
      for (int c = 0; c < 4; ++c) {
        int kk = 4 * c + 2 * hg;
        a2A[c].x = A2p[ln * 16 + kk]; a2A[c].y = A2p[ln * 16 + kk + 1];
        a1B[c].x = A1[kk * 16 + ln];  a1B[c].y = A1[(kk + 1) * 16 + ln];
        n1B[c].x = N1[kk * 16 + ln];  n1B[c].y = N1[(kk + 1) * 16 + ln];
      }
      v8f n2c;
#pragma unroll
      for (int r = 0; r < 8; ++r) n2c[r] = N2p[(mb + r) * 16 + ln];
      v8f accA = {0.f, 0.f, 0.f, 0.f, 0.f, 0.f, 0.f, 0.f};
      v8f accY = {0.f, 0.f, 0.f, 0.f, 0.f, 0.f, 0.f, 0.f};
#pragma unroll
      for (int c = 0; c < 4; ++c) {
        accA = WMMA4(a2A[c], a1B[c], accA);
        accY = WMMA4(a2A[c], n1B[c], accY);
      }
#pragma unroll
      for (int r = 0; r < 8; ++r) {
        A1[(mb + r) * 16 + ln] = accA[r];
        Ys[(mb + r) * 16 + ln] = accY[r];
      }
      LDS_FENCE();
      v2f yA[4];
#pragma unroll
      for (int c = 0; c < 4; ++c) {
        int kk = 4 * c + 2 * hg;
        yA[c].x = Ys[ln * 16 + kk]; yA[c].y = Ys[ln * 16 + kk + 1];
      }
      v8f accN = n2c;
#pragma unroll
      for (int c = 0; c < 4; ++c) accN = WMMA4(yA[c], a2A[c], accN);
#pragma unroll
      for (int r = 0; r < 8; ++r) N1[(mb + r) * 16 + ln] = accN[r];
      LDS_FENCE();
    }
    for (int i = lane; i < 81; i += 32)
      cov_out[(size_t)b * 81 + i] = N1[(i / 9) * 16 + (i % 9)];
  }
}

extern "C" void kernel_launch(void* const* d_in, const int* in_sizes, int n_in,
                              void* d_out, int out_size, void* d_ws, size_t ws_size,
                              hipStream_t stream) {
  (void)in_sizes; (void)n_in; (void)out_size; (void)d_ws; (void)ws_size;
  const float* dt   = (const float*)d_in[0];
  const float* gyro = (const float*)d_in[1];
  const float* acc  = (const float*)d_in[2];
  const float* pos0 = (const float*)d_in[3];
  const float* vel0 = (const float*)d_in[4];
  float* out = (float*)d_out;
  float* quat_out = out;                                        // 32*8192*4
  float* vel_out  = quat_out + (size_t)B_DIM * F_DIM * 4;       // 32*8192*3
  float* pos_out  = vel_out + (size_t)B_DIM * F_DIM * 3;        // 32*8192*3
  float* cov_out  = pos_out + (size_t)B_DIM * F_DIM * 3;        // 32*81

  k_integrate<<<dim3(B_DIM), dim3(256), 0, stream>>>(
      dt, gyro, acc, pos0, vel0, quat_out, vel_out, pos_out);
  k_cov<<<dim3(B_DIM), dim3(256), 0, stream>>>(
      dt, gyro, acc, quat_out, cov_out);
}